// GCNConv2_63788854280594
// MI455X (gfx1250) — compile-verified
//
#include <hip/hip_runtime.h>

typedef __attribute__((ext_vector_type(2))) float v2f;
typedef __attribute__((ext_vector_type(8))) float v8f;

#define N_NODES 100000
#define IN_CH   256
#define OUT_CH  128
#define N_EDGES 3200000

// LDS x-tile: 16 rows x 256 cols, row stride padded to 258 floats.
//  - bank for lane l15 reading (k): (l15*258 + k) % 64 = (2*l15 + k) % 64
//    -> 16 lanes per half hit 16 distinct bank pairs (conflict-free b64).
//  - 258*4 = 1032 bytes, divisible by 8 -> every float2/b64 access stays
//    8-byte aligned. Total 16*258*4 = 16.5 KB of the WGP's 320 KB LDS.
#define XS_STRIDE 258

// ---------------------------------------------------------------------------
// Kernel 1: h = x @ W + b   via V_WMMA_F32_16X16X4_F32
//   grid.x = N_NODES/16 = 6250 blocks of 256 threads (8 waves).
//   Block b owns M-tile b (16 rows of x, staged once in LDS);
//   wave w owns N-tile w (16 of 128 cols). 64 K-steps of 4.
//
// Fragment layouts per CDNA5 ISA 7.12.2 (wave32):
//   A 16x4 f32 : lane L<16 holds A[M=L][K=0..1]; lane L>=16 holds K=2..3
//                -> one ds_load_b64 (float2) per lane per step
//   B 4x16 f32 : mirrored row-striped layout -> two b32 loads (W is 128 KB,
//                L2/WGP$-resident and broadcast across all 6250 blocks)
//   C/D 16x16  : v8f; element r of lane L is (M = r + 8*(L>=16), N = L&15)
// ---------------------------------------------------------------------------
__global__ __launch_bounds__(256) void gemm_wmma_f32_kernel(
    const float* __restrict__ x, const float* __restrict__ W,
    const float* __restrict__ bias, float* __restrict__ h)
{
  __shared__ float xs[16 * XS_STRIDE];

  const int tileM = blockIdx.x << 4;          // 16 rows per block
  const int tid   = threadIdx.x;

  // --- Stage the 16x256 x-tile into LDS (float2 granularity, coalesced) ---
  // 16 rows * 128 float2 = 2048 pairs; 256 threads -> 8 pairs each.
#pragma unroll
  for (int j = 0; j < 8; ++j) {
    const int pairIdx = tid + (j << 8);       // 0..2047
    const int row = pairIdx >> 7;             // 0..15
    const int cp  = (pairIdx & 127) << 1;     // even column 0..254
    const v2f xv = *(const v2f*)(x + (size_t)(tileM + row) * IN_CH + cp);
    *(v2f*)(xs + row * XS_STRIDE + cp) = xv;  // 8B-aligned ds_store_b64
  }
  __syncthreads();

  const int wave  = tid >> 5;                 // 0..7 -> N tile
  const int lane  = tid & 31;
  const int tileN = wave << 4;
  const int half  = lane >> 4;                // 0 or 1 (K pair select)
  const int l15   = lane & 15;
  const int koff  = half << 1;                // 0 or 2

  // A from LDS: xs[l15][k + koff + {0,1}]
  const float* arow = xs + l15 * XS_STRIDE + koff;
  // B from global: W[k + koff + {0,1}][tileN + l15]
  const float* bcol = W + (size_t)koff * OUT_CH + tileN + l15;

  v8f acc = {};
#pragma unroll 8
  for (int k = 0; k < IN_CH; k += 4) {
    v2f a = *(const v2f*)(arow + k);          // ds_load_b64, conflict-free
    v2f b;
    b.x = bcol[(size_t)k * OUT_CH];
    b.y = bcol[(size_t)k * OUT_CH + OUT_CH];
    // (neg_a, A, neg_b, B, c_mod, C, reuse_a, reuse_b)
    acc = __builtin_amdgcn_wmma_f32_16x16x4_f32(
        false, a, false, b, (short)0, acc, false, false);
  }

  // Fused bias add + store of the 16x16 tile (h stays RT -> L2-resident).
  const float bv = bias[tileN + l15];
  float* hrow = h + (size_t)(tileM + half * 8) * OUT_CH + tileN + l15;
#pragma unroll
  for (int r = 0; r < 8; ++r) {
    hrow[(size_t)r * OUT_CH] = acc[r] + bv;
  }
}

// ---------------------------------------------------------------------------
// Kernel 2: out[row[e]] += val[e] * h[col[e]]   (wave-per-edge)
//   32 lanes x float4 = 128 channels per edge. h (51.2 MB) and out (51.2 MB)
//   both fit in MI455X's 192 MB L2, so the random gathers and the 410M
//   global_atomic_add_f32 ops resolve in L2. The edge lists (38 MB) are
//   streamed exactly once -> non-temporal loads so they don't evict h/out.
// ---------------------------------------------------------------------------
__global__ __launch_bounds__(256) void edge_scatter_kernel(
    const int* __restrict__ erow, const int* __restrict__ ecol,
    const float* __restrict__ eval, const float* __restrict__ h,
    float* __restrict__ out)
{
  const int e    = (int)((blockIdx.x * blockDim.x + threadIdx.x) >> 5);
  const int lane = threadIdx.x & 31;
  if (e >= N_EDGES) return;

  const int   r = __builtin_nontemporal_load(erow + e);
  const int   c = __builtin_nontemporal_load(ecol + e);
  const float v = __builtin_nontemporal_load(eval + e);

  const float4 hv = *(const float4*)(h + (size_t)c * OUT_CH + lane * 4);
  float* o = out + (size_t)r * OUT_CH + lane * 4;
  atomicAdd(o + 0, v * hv.x);
  atomicAdd(o + 1, v * hv.y);
  atomicAdd(o + 2, v * hv.z);
  atomicAdd(o + 3, v * hv.w);
}

extern "C" void kernel_launch(void* const* d_in, const int* in_sizes, int n_in,
                              void* d_out, int out_size, void* d_ws, size_t ws_size,
                              hipStream_t stream) {
  const float* x    = (const float*)d_in[0];   // [100000, 256]
  const float* W    = (const float*)d_in[1];   // [256, 128]
  const float* bias = (const float*)d_in[2];   // [128]
  const int*   erow = (const int*)d_in[3];     // [3.2M]
  const int*   ecol = (const int*)d_in[4];     // [3.2M]
  const float* eval = (const float*)d_in[5];   // [3.2M]
  float* out = (float*)d_out;                  // [100000, 128]
  float* h   = (float*)d_ws;                   // scratch: 51.2 MB

  (void)in_sizes; (void)n_in; (void)ws_size;

  // Zero the scatter-add destination (graph-capture-safe memset node).
  hipMemsetAsync(d_out, 0, (size_t)out_size * sizeof(float), stream);

  // GEMM: one block per 16-row M tile; 8 waves cover all 128 output columns.
  gemm_wmma_f32_kernel<<<N_NODES / 16, 256, 0, stream>>>(x, W, bias, h);

  // Scatter: one wave per edge, 8 waves per block.
  edge_scatter_kernel<<<(N_EDGES + 7) / 8, 256, 0, stream>>>(erow, ecol, eval, h, out);
}